// GATv2Block_66211215835636
// MI455X (gfx1250) — compile-verified
//
#include <hip/hip_runtime.h>
#include <math.h>

typedef float v2f __attribute__((ext_vector_type(2)));
typedef float v8f __attribute__((ext_vector_type(8)));

__device__ __forceinline__ void atomicMaxF32(float* addr, float val) {
  int* ai = reinterpret_cast<int*>(addr);
  int cur = *ai;
  while (__int_as_float(cur) < val) {
    int assumed = cur;
    cur = atomicCAS(ai, assumed, __float_as_int(val));
    if (cur == assumed) break;
  }
}

// ---------------- fill ----------------
__global__ void fill_f32(float* __restrict__ p, float v, int n) {
  int t = blockIdx.x * blockDim.x + threadIdx.x;
  if (t < n) p[t] = v;
}

// ---------------- GEMM: C[n x 128] = A[n x 128] @ W[128 x 128] + bias ----
// block = 256 threads = 8 wave32; block covers 16 rows x 128 cols,
// each wave computes one 16x16 tile via 32 x v_wmma_f32_16x16x4_f32.
__global__ __launch_bounds__(256) void gemm_xform(
    const float* __restrict__ A, const float* __restrict__ W,
    const float* __restrict__ bias, float* __restrict__ C, int n)
{
  __shared__ float As[16][132];          // pad 132 -> conflict-free frag reads
  const int t    = threadIdx.x;
  const int wave = t >> 5;
  const int lane = t & 31;
  const int row0 = blockIdx.x << 4;

  // cooperative load of the 16x128 A tile (float4 per thread, 2 iters)
  #pragma unroll
  for (int i = 0; i < 2; ++i) {
    int id = t + (i << 8);               // 0..511 float4 slots
    int r  = id >> 5;
    int c4 = id & 31;
    int gr = row0 + r;
    float4 v = make_float4(0.f, 0.f, 0.f, 0.f);
    if (gr < n) v = *reinterpret_cast<const float4*>(&A[(long)gr * 128 + (c4 << 2)]);
    *reinterpret_cast<float4*>(&As[r][c4 << 2]) = v;
  }
  __syncthreads();

  const int colBase = wave << 4;
  const int half    = lane >> 4;         // 0: K=0,1  1: K=2,3
  const int lrow    = lane & 15;
  const float* Wc   = W + colBase + lrow;

  v8f acc = {};
  #pragma unroll
  for (int k = 0; k < 128; k += 4) {
    const int kk = k + (half << 1);
    v2f a; a.x = As[lrow][kk]; a.y = As[lrow][kk + 1];     // A frag (ISA layout)
    v2f b; b.x = Wc[kk << 7];  b.y = Wc[(kk + 1) << 7];    // B frag (row-striped)
    acc = __builtin_amdgcn_wmma_f32_16x16x4_f32(
        false, a, false, b, (short)0, acc, false, false);
  }

  const float bc = bias[colBase + lrow];
  #pragma unroll
  for (int v = 0; v < 8; ++v) {
    const int gr = row0 + v + (half << 3);                 // C/D layout
    if (gr < n) C[(long)gr * 128 + colBase + lrow] = acc[v] + bc;
  }
}

// ------------- pass 1: per-edge logits + segment max over dst -------------
// one wave32 per edge; lane l owns channels l, l+32, l+64, l+96
template<int HH>
__global__ __launch_bounds__(256) void edge_logits(
    const float* __restrict__ xl, const float* __restrict__ xr,
    const float* __restrict__ att, const int* __restrict__ esrc,
    const int* __restrict__ edst, int E0, int Etot,
    float* __restrict__ logits, float* __restrict__ mmax)
{
  const int e = (blockIdx.x << 3) + (threadIdx.x >> 5);
  if (e >= Etot) return;
  const int lane = threadIdx.x & 31;
  int s, d;
  if (e < E0) { s = esrc[e]; d = edst[e]; } else { s = e - E0; d = s; }
  const float* pl = xl + (long)s * 128;
  const float* pr = xr + (long)d * 128;

  float sums[HH];
  #pragma unroll
  for (int h = 0; h < HH; ++h) sums[h] = 0.f;
  #pragma unroll
  for (int j = 0; j < 4; ++j) {
    const int h = (32 * j) / (128 / HH);   // compile-time head per slot
    const int c = lane + (j << 5);
    float v = pl[c] + pr[c];
    v = (v > 0.f) ? v : 0.2f * v;          // LeakyReLU(0.2)
    sums[h] += v * att[c];
  }
  #pragma unroll
  for (int h = 0; h < HH; ++h) {
    float v = sums[h];
    #pragma unroll
    for (int off = 16; off > 0; off >>= 1) v += __shfl_xor(v, off, 32);
    sums[h] = v;
  }
  if (lane == 0) {
    #pragma unroll
    for (int h = 0; h < HH; ++h) {
      logits[(long)e * HH + h] = sums[h];
      atomicMaxF32(&mmax[(long)d * HH + h], sums[h]);
    }
  }
}

// ------------- pass 2: p = exp(logit - max[dst]); denom[dst] += p ---------
__global__ void edge_exp_denom(float* __restrict__ logits,
                               const int* __restrict__ edst,
                               int E0, int Etot,
                               const float* __restrict__ mmax,
                               float* __restrict__ denom, int HH)
{
  const int t = blockIdx.x * blockDim.x + threadIdx.x;
  const int total = Etot * HH;
  if (t >= total) return;
  const int e = t / HH;
  const int h = t - e * HH;
  const int d = (e < E0) ? edst[e] : (e - E0);
  const float p = expf(logits[t] - mmax[d * HH + h]);
  logits[t] = p;                                   // in-place p
  atomicAdd(&denom[d * HH + h], p);
}

// ------------- pass 3: agg[dst] += alpha * xl[src] ------------------------
template<int HH>
__global__ __launch_bounds__(256) void edge_scatter(
    const float* __restrict__ xl, const float* __restrict__ p,
    const float* __restrict__ denom, const int* __restrict__ esrc,
    const int* __restrict__ edst, int E0, int Etot, float* __restrict__ agg)
{
  const int e = (blockIdx.x << 3) + (threadIdx.x >> 5);
  if (e >= Etot) return;
  const int lane = threadIdx.x & 31;
  int s, d;
  if (e < E0) { s = esrc[e]; d = edst[e]; } else { s = e - E0; d = s; }
  float alpha[HH];
  #pragma unroll
  for (int h = 0; h < HH; ++h)
    alpha[h] = p[(long)e * HH + h] / denom[(long)d * HH + h];
  const float* px = xl + (long)s * 128;
  float* pa = agg + (long)d * 128;
  #pragma unroll
  for (int j = 0; j < 4; ++j) {
    const int h = (32 * j) / (128 / HH);
    const int c = lane + (j << 5);
    atomicAdd(&pa[c], alpha[h] * px[c]);           // global_atomic_add_f32
  }
}

// ------------- bias (+ optional ReLU), in place ---------------------------
__global__ void bias_act(float* __restrict__ buf, const float* __restrict__ bias,
                         int total, int do_relu)
{
  int t = blockIdx.x * blockDim.x + threadIdx.x;
  if (t >= total) return;
  float v = buf[t] + bias[t & 127];
  if (do_relu) v = fmaxf(v, 0.f);
  buf[t] = v;
}

extern "C" void kernel_launch(void* const* d_in, const int* in_sizes, int n_in,
                              void* d_out, int out_size, void* d_ws, size_t ws_size,
                              hipStream_t stream)
{
  (void)n_in; (void)out_size; (void)ws_size;
  const float* x     = (const float*)d_in[0];
  const int*   ei    = (const int*)d_in[1];
  const float* W1l   = (const float*)d_in[2];
  const float* b1l   = (const float*)d_in[3];
  const float* W1r   = (const float*)d_in[4];
  const float* b1r   = (const float*)d_in[5];
  const float* att1  = (const float*)d_in[6];
  const float* bias1 = (const float*)d_in[7];
  const float* W2l   = (const float*)d_in[8];
  const float* b2l   = (const float*)d_in[9];
  const float* W2r   = (const float*)d_in[10];
  const float* b2r   = (const float*)d_in[11];
  const float* att2  = (const float*)d_in[12];
  const float* bias2 = (const float*)d_in[13];

  const int N    = in_sizes[0] / 128;
  const int E0   = in_sizes[1] / 2;
  const int Etot = E0 + N;
  const int* esrc = ei;
  const int* edst = ei + E0;

  const long NF = (long)N * 128;
  float* xl   = (float*)d_ws;            // N*128
  float* xr   = xl + NF;                 // N*128
  float* agg1 = xr + NF;                 // N*128 (becomes hidden after relu)
  float* plog = agg1 + NF;               // Etot*4 (logits, then p)
  float* mmax = plog + (long)Etot * 4;   // N*4
  float* den  = mmax + (long)N * 4;      // N*4
  float* out  = (float*)d_out;

  const int B      = 256;
  const int gN128  = (int)((NF + B - 1) / B);
  const int gNH4   = (N * 4 + B - 1) / B;
  const int gN1    = (N + B - 1) / B;
  const int gGemm  = (N + 15) / 16;
  const int gEdgeW = (Etot + 7) / 8;

  // ---------------- conv1 (H=4) ----------------
  fill_f32<<<gNH4, B, 0, stream>>>(mmax, -INFINITY, N * 4);
  fill_f32<<<gNH4, B, 0, stream>>>(den, 0.f, N * 4);
  fill_f32<<<gN128, B, 0, stream>>>(agg1, 0.f, (int)NF);
  gemm_xform<<<gGemm, B, 0, stream>>>(x, W1l, b1l, xl, N);
  gemm_xform<<<gGemm, B, 0, stream>>>(x, W1r, b1r, xr, N);
  edge_logits<4><<<gEdgeW, B, 0, stream>>>(xl, xr, att1, esrc, edst, E0, Etot, plog, mmax);
  edge_exp_denom<<<(Etot * 4 + B - 1) / B, B, 0, stream>>>(plog, edst, E0, Etot, mmax, den, 4);
  edge_scatter<4><<<gEdgeW, B, 0, stream>>>(xl, plog, den, esrc, edst, E0, Etot, agg1);
  bias_act<<<gN128, B, 0, stream>>>(agg1, bias1, (int)NF, 1);   // relu -> hidden

  // ---------------- conv2 (H=1) ----------------
  fill_f32<<<gN1, B, 0, stream>>>(mmax, -INFINITY, N);
  fill_f32<<<gN1, B, 0, stream>>>(den, 0.f, N);
  fill_f32<<<gN128, B, 0, stream>>>(out, 0.f, (int)NF);
  gemm_xform<<<gGemm, B, 0, stream>>>(agg1, W2l, b2l, xl, N);
  gemm_xform<<<gGemm, B, 0, stream>>>(agg1, W2r, b2r, xr, N);
  edge_logits<1><<<gEdgeW, B, 0, stream>>>(xl, xr, att2, esrc, edst, E0, Etot, plog, mmax);
  edge_exp_denom<<<(Etot + B - 1) / B, B, 0, stream>>>(plog, edst, E0, Etot, mmax, den, 1);
  edge_scatter<1><<<gEdgeW, B, 0, stream>>>(xl, plog, den, esrc, edst, E0, Etot, out);
  bias_act<<<gN128, B, 0, stream>>>(out, bias2, (int)NF, 0);
}